// SubTrajectoryBalanceLoss_87806311400229
// MI455X (gfx1250) — compile-verified
//
#include <hip/hip_runtime.h>

// SubTrajectoryBalanceLoss — MI455X (gfx1250, wave32)
//
// residual[b,t] = (log_z - log_R[b]) + sum_{u>=t} (log_pf[b,u] - log_pb[b,u])
// out = (1/(B*L)) * sum_b sum_t 0.9^(L-1-t) * residual^2
//
// Memory-bound: weights underflow to exactly 0.0f beyond ~1024 trailing
// columns, so we stream row tiles from the end and stop when the tile weight
// scale hits 0. One wave per row; float4 loads (512B/wave/tile per array).

#define LAMBDA_SUB 0.9f

typedef __attribute__((ext_vector_type(2))) float v2f;
typedef __attribute__((ext_vector_type(8))) float v8f;

__global__ __launch_bounds__(256) void stb_rowscan_kernel(
    const float* __restrict__ pf, const float* __restrict__ pb,
    const float* __restrict__ logR, const float* __restrict__ logZ,
    float* __restrict__ rowsum, int B, int L)
{
    const int lane = threadIdx.x & 31;
    const int wave = threadIdx.x >> 5;
    const int row  = blockIdx.x * (blockDim.x >> 5) + wave;
    if (row >= B) return;

    const int K = L >> 7;  // tiles of 128 columns (L is a multiple of 128)
    const float4* __restrict__ pf4 = (const float4*)(pf + (size_t)row * (size_t)L);
    const float4* __restrict__ pb4 = (const float4*)(pb + (size_t)row * (size_t)L);

    const float c = logZ[0] - logR[row];

    // Weight bases at the last tile (tileScale == 1): element (lane, j) sits at
    // t = L-128 + 4*lane + j, exponent = 127 - 4*lane - j.
    const float w3 = __powf(LAMBDA_SUB, (float)(124 - 4 * lane));
    const float w2 = w3 * LAMBDA_SUB;
    const float w1 = w2 * LAMBDA_SUB;
    const float w0 = w1 * LAMBDA_SUB;
    const float lam128 = __powf(LAMBDA_SUB, 128.0f);  // per-tile decay ~1.39e-6

    float carry = 0.0f;   // suffix sum of d over all tiles already processed
    float acc   = 0.0f;   // sum_t w_t * r_t^2 for this row
    float scale = 1.0f;   // 0.9^(128 * tiles_processed)

    for (int k = K - 1; k >= 0; --k) {
        if (k > 0) {  // gfx1250 global_prefetch_b8 for the next tile
            __builtin_prefetch(&pf4[((k - 1) << 5) + lane], 0, 0);
            __builtin_prefetch(&pb4[((k - 1) << 5) + lane], 0, 0);
        }
        const float4 a = pf4[(k << 5) + lane];
        const float4 b = pb4[(k << 5) + lane];
        const float d0 = a.x - b.x, d1 = a.y - b.y;
        const float d2 = a.z - b.z, d3 = a.w - b.w;

        // local reverse (suffix) scan over this lane's 4 elements
        const float s3 = d3;
        const float s2 = d2 + s3;
        const float s1 = d1 + s2;
        const float s0 = d0 + s1;

        // wave32 inclusive suffix scan of per-lane totals (s0)
        float incl = s0;
        #pragma unroll
        for (int off = 1; off < 32; off <<= 1) {
            const float t = __shfl_down(incl, off, 32);
            incl += (lane + off < 32) ? t : 0.0f;
        }
        const float tileTotal = __shfl(incl, 0, 32);  // sum of whole tile
        const float excl = incl - s0;                 // lanes strictly above

        const float base = excl + carry + c;
        const float r0 = s0 + base, r1 = s1 + base;
        const float r2 = s2 + base, r3 = s3 + base;

        acc = fmaf(w0 * scale, r0 * r0, acc);
        acc = fmaf(w1 * scale, r1 * r1, acc);
        acc = fmaf(w2 * scale, r2 * r2, acc);
        acc = fmaf(w3 * scale, r3 * r3, acc);

        carry += tileTotal;
        scale *= lam128;
        if (scale == 0.0f) break;  // all remaining weights are exactly 0.0f
    }

    // wave reduction of acc
    #pragma unroll
    for (int off = 16; off > 0; off >>= 1)
        acc += __shfl_down(acc, off, 32);
    if (lane == 0) rowsum[row] = acc;
}

// Stage 2: reduce B row-partials with V_WMMA_F32_16X16X4_F32.
// A (16x4 f32) = 64 partials per chunk; B = all-ones; C accumulates 16x16 f32
// whose every column equals the per-M chunk sums. Launch exactly one wave32,
// no divergence -> EXEC all-ones as WMMA requires.
__global__ __launch_bounds__(32) void stb_reduce_wmma_kernel(
    const float* __restrict__ rowsum, float* __restrict__ out, int B, int L)
{
    const int lane = threadIdx.x & 31;
    const int m    = lane & 15;          // A-matrix row M
    const int koff = (lane >> 4) << 1;   // lanes 0-15 hold K=0,1; lanes 16-31 K=2,3

    v2f bmat;
    bmat[0] = 1.0f;
    bmat[1] = 1.0f;
    v8f cacc = {};

    const int chunks = B >> 6;           // 64 values per WMMA
    for (int ch = 0; ch < chunks; ++ch) {
        const float* p = rowsum + (ch << 6) + (m << 2) + koff;  // element (m, k)
        v2f amat;
        amat[0] = p[0];
        amat[1] = p[1];
        cacc = __builtin_amdgcn_wmma_f32_16x16x4_f32(
            /*neg_a=*/false, amat, /*neg_b=*/false, bmat,
            /*c_mod=*/(short)0, cacc, /*reuse_a=*/false, /*reuse_b=*/false);
    }

    // column 0 of C: rows M=0..7 live in lane 0 (VGPRs 0..7), M=8..15 in lane 16
    float colsum = cacc[0] + cacc[1] + cacc[2] + cacc[3]
                 + cacc[4] + cacc[5] + cacc[6] + cacc[7];

    // scalar tail for B % 64 (zero for B=4096, kept for generality)
    float tail = 0.0f;
    for (int i = (chunks << 6) + lane; i < B; i += 32) tail += rowsum[i];
    #pragma unroll
    for (int off = 16; off > 0; off >>= 1)
        tail += __shfl_down(tail, off, 32);

    const float total = __shfl(colsum, 0, 32) + __shfl(colsum, 16, 32)
                      + __shfl(tail, 0, 32);
    if (lane == 0)
        out[0] = total / ((float)B * (float)L);
}

extern "C" void kernel_launch(void* const* d_in, const int* in_sizes, int n_in,
                              void* d_out, int out_size, void* d_ws, size_t ws_size,
                              hipStream_t stream) {
    const float* pf   = (const float*)d_in[0];  // [B, L]
    const float* pb   = (const float*)d_in[1];  // [B, L]
    const float* logR = (const float*)d_in[2];  // [B]
    const float* logZ = (const float*)d_in[3];  // scalar

    const int B = in_sizes[2];
    const int L = in_sizes[0] / B;

    float* rowsum = (float*)d_ws;  // B floats of scratch

    const int rowsPerBlock = 8;    // 8 wave32 per 256-thread block
    dim3 grid((B + rowsPerBlock - 1) / rowsPerBlock);
    stb_rowscan_kernel<<<grid, 256, 0, stream>>>(pf, pb, logR, logZ, rowsum, B, L);

    stb_reduce_wmma_kernel<<<1, 32, 0, stream>>>(rowsum, (float*)d_out, B, L);
}